// TreeGNNLayer_69509750718668
// MI455X (gfx1250) — compile-verified
//
#include <hip/hip_runtime.h>
#include <hip/hip_bf16.h>
#include <math.h>
#include <stdint.h>

#define N_NODES 50000
#define DIM 128
#define N_EDGES 800000
#define NEG_SLOPE 0.2f
#define LN_EPS 1e-5f

typedef float v2f __attribute__((ext_vector_type(2)));
typedef float v8f __attribute__((ext_vector_type(8)));
typedef unsigned u32x4 __attribute__((ext_vector_type(4)));
typedef int i32x4 __attribute__((ext_vector_type(4)));
typedef int i32x8 __attribute__((ext_vector_type(8)));

// ---- order-preserving float<->uint mapping for atomicMax on floats ----
__device__ __forceinline__ unsigned f2ord(float f) {
    unsigned u = __float_as_uint(f);
    return (u & 0x80000000u) ? ~u : (u | 0x80000000u);
}
__device__ __forceinline__ float ord2f(unsigned u) {
    return (u & 0x80000000u) ? __uint_as_float(u ^ 0x80000000u)
                             : __uint_as_float(~u);
}

// ---------------------------------------------------------------------
// Kernel 0: zero-init workspace accumulators
// ---------------------------------------------------------------------
__global__ __launch_bounds__(256) void k_init(float* __restrict__ agg,
                                              unsigned* __restrict__ seg_ord,
                                              float* __restrict__ seg_sum) {
    int i = blockIdx.x * 256 + threadIdx.x;
    if (i < N_NODES * DIM) agg[i] = 0.0f;
    if (i < N_NODES) { seg_ord[i] = 0u; seg_sum[i] = 0.0f; }  // ord 0 == minimum
}

// ---------------------------------------------------------------------
// Kernel 1: x_l = x@W_l + b_l ; x_r = x@W_r + b_r
//   - W staged into LDS by the Tensor Data Mover (tensor_load_to_lds),
//     TENSORcnt-waited by the issuing wave, then shared across 8 waves.
//   - asm memory-clobber makes the TDM's LDS write visible to the compiler
//     so the ds_load reads of sW are not folded to poison.
//   - Matrix math via V_WMMA_F32_16X16X4_F32 (exact fp32 parity with ref).
//   grid = (N/16, 2); block = 256 (8 waves). Wave w -> output cols [16w,16w+16).
// ---------------------------------------------------------------------
__global__ __launch_bounds__(256) void k_gemm_xlxr(const float* __restrict__ x,
                                                   const float* __restrict__ Wl,
                                                   const float* __restrict__ bl,
                                                   const float* __restrict__ Wr,
                                                   const float* __restrict__ br,
                                                   float* __restrict__ x_l,
                                                   float* __restrict__ x_r) {
    __shared__ float sW[DIM * DIM];   // 64 KB

    const int wave = threadIdx.x >> 5;
    const int lane = threadIdx.x & 31;
    const int m0   = blockIdx.x * 16;
    const bool right = (blockIdx.y != 0);
    const float* __restrict__ W    = right ? Wr : Wl;
    const float* __restrict__ bias = right ? br : bl;
    float* __restrict__ out        = right ? x_r : x_l;

#if __has_builtin(__builtin_amdgcn_tensor_load_to_lds)
    if (threadIdx.x < 32) {   // wave 0 issues one TDM descriptor for all of W
        unsigned long long wa = (unsigned long long)(const void*)W;
        // flat-address low 32 bits == LDS byte offset (ISA aperture mapping)
        unsigned ldsOff = (unsigned)(uintptr_t)(void*)sW;
        u32x4 g0;
        g0[0] = 1u;                                    // count=1, user desc
        g0[1] = ldsOff;                                // lds_addr (bytes)
        g0[2] = (unsigned)(wa & 0xFFFFFFFFull);        // global_addr[31:0]
        g0[3] = (unsigned)((wa >> 32) & 0x01FFFFFFull) // global_addr[56:32]
              | (2u << 30);                            // type = 2 ("image")
        i32x8 g1;
        g1[0] = 0x20000;              // data_size = 2 (4 bytes); no iterate/pad
        g1[1] = (int)(128u << 16);    // tensor_dim0 = 128 (lo16 in [31:16])
        g1[2] = (int)(128u << 16);    // tensor_dim1 = 128
        g1[3] = (int)(128u << 16);    // tile_dim0 = 128
        g1[4] = 128;                  // tile_dim1 = 128, tile_dim2 = 0
        g1[5] = 128;                  // tensor_dim0_stride = 128 (lo32)
        g1[6] = 0;                    // stride hi / dim1_stride (unused, 2D)
        g1[7] = 0;
        i32x4 z4 = {0, 0, 0, 0};      // groups 2/3 unused for 2D tile
        i32x8 z8 = {0, 0, 0, 0, 0, 0, 0, 0};
        __builtin_amdgcn_tensor_load_to_lds(g0, g1, z4, z4, z8, 0);
        __builtin_amdgcn_s_wait_tensorcnt(0);
    }
    // Compiler-visibility fence: the TDM wrote sW behind the compiler's back.
    asm volatile("" : : "r"(&sW[0]) : "memory");
    __syncthreads();
#else
    for (int i = threadIdx.x; i < DIM * DIM / 4; i += 256)
        ((float4*)sW)[i] = ((const float4*)W)[i];
    __syncthreads();
#endif

    const int n0   = wave * 16;
    const int col  = n0 + (lane & 15);
    const int rowA = m0 + (lane & 15);
    const int koff = (lane >> 4) * 2;   // 0 or 2

    v8f acc = {};
#pragma unroll 4
    for (int k = 0; k < DIM; k += 4) {
        v2f a, b;
        const float* ap = x + rowA * DIM + k + koff;
        a.x = ap[0];
        a.y = ap[1];
        b.x = sW[(k + koff    ) * DIM + col];   // ds_load_b32 from staged W
        b.y = sW[(k + koff + 1) * DIM + col];
        acc = __builtin_amdgcn_wmma_f32_16x16x4_f32(false, a, false, b,
                                                    (short)0, acc, false, false);
    }
    const float bv = bias[col];
    const int rbase = m0 + (lane >> 4) * 8;
#pragma unroll
    for (int i = 0; i < 8; ++i)
        out[(rbase + i) * DIM + col] = acc[i] + bv;
}

// ---------------------------------------------------------------------
// Kernel 2: per-edge attention logit + segment max (one wave per edge)
// ---------------------------------------------------------------------
__global__ __launch_bounds__(256) void k_edge_logits(const int* __restrict__ ei,
                                                     const float* __restrict__ edge_attr,
                                                     const float* __restrict__ W_e,
                                                     const float* __restrict__ att,
                                                     const float* __restrict__ x_l,
                                                     const float* __restrict__ x_r,
                                                     float* __restrict__ alpha,
                                                     unsigned* __restrict__ seg_ord) {
    const int e = blockIdx.x * 8 + (threadIdx.x >> 5);
    if (e >= N_EDGES) return;
    const int lane = threadIdx.x & 31;
    const int src = ei[e];
    const int dst = ei[N_EDGES + e];
    const float ea0 = edge_attr[e * 3 + 0];
    const float ea1 = edge_attr[e * 3 + 1];
    const float ea2 = edge_attr[e * 3 + 2];

    const float4 a  = ((const float4*)(x_l + src * DIM))[lane];
    const float4 b  = ((const float4*)(x_r + dst * DIM))[lane];
    const float4 w0 = ((const float4*)(W_e          ))[lane];
    const float4 w1 = ((const float4*)(W_e + DIM    ))[lane];
    const float4 w2 = ((const float4*)(W_e + 2 * DIM))[lane];
    const float4 at = ((const float4*)att)[lane];

    float dot = 0.0f;
    {
        float m;
        m = a.x + b.x + ea0 * w0.x + ea1 * w1.x + ea2 * w2.x;
        m = (m > 0.0f) ? m : NEG_SLOPE * m;  dot += m * at.x;
        m = a.y + b.y + ea0 * w0.y + ea1 * w1.y + ea2 * w2.y;
        m = (m > 0.0f) ? m : NEG_SLOPE * m;  dot += m * at.y;
        m = a.z + b.z + ea0 * w0.z + ea1 * w1.z + ea2 * w2.z;
        m = (m > 0.0f) ? m : NEG_SLOPE * m;  dot += m * at.z;
        m = a.w + b.w + ea0 * w0.w + ea1 * w1.w + ea2 * w2.w;
        m = (m > 0.0f) ? m : NEG_SLOPE * m;  dot += m * at.w;
    }
#pragma unroll
    for (int off = 16; off > 0; off >>= 1) dot += __shfl_xor(dot, off, 32);

    if (lane == 0) {
        alpha[e] = dot;
        atomicMax(seg_ord + dst, f2ord(dot));
    }
}

// ---------------------------------------------------------------------
// Kernel 3: ex = exp(alpha - segmax[dst]); segsum[dst] += ex (1 thread/edge)
// ---------------------------------------------------------------------
__global__ __launch_bounds__(256) void k_edge_exp(const int* __restrict__ ei,
                                                  float* __restrict__ alpha,  // in: logit, out: ex
                                                  const unsigned* __restrict__ seg_ord,
                                                  float* __restrict__ seg_sum) {
    const int e = blockIdx.x * 256 + threadIdx.x;
    if (e >= N_EDGES) return;
    const int dst = ei[N_EDGES + e];
    const float m = ord2f(seg_ord[dst]);          // finite for any dst with an edge
    const float ex = __expf(alpha[e] - m);
    alpha[e] = ex;
    atomicAdd(seg_sum + dst, ex);
}

// ---------------------------------------------------------------------
// Kernel 4: scatter-add msgs: agg[dst] += x_l[src] * (ex / (segsum+eps))
//   one wave per edge, 4 features/lane, atomic f32 adds into L2
// ---------------------------------------------------------------------
__global__ __launch_bounds__(256) void k_edge_scatter(const int* __restrict__ ei,
                                                      const float* __restrict__ ex,
                                                      const float* __restrict__ seg_sum,
                                                      const float* __restrict__ x_l,
                                                      float* __restrict__ agg) {
    const int e = blockIdx.x * 8 + (threadIdx.x >> 5);
    if (e >= N_EDGES) return;
    const int lane = threadIdx.x & 31;
    const int src = ei[e];
    const int dst = ei[N_EDGES + e];
    const float w = ex[e] / (seg_sum[dst] + 1e-16f);

    const float4 v = ((const float4*)(x_l + src * DIM))[lane];
    float* dp = agg + dst * DIM + lane * 4;
    atomicAdd(dp + 0, v.x * w);
    atomicAdd(dp + 1, v.y * w);
    atomicAdd(dp + 2, v.z * w);
    atomicAdd(dp + 3, v.w * w);
}

// ---------------------------------------------------------------------
// Kernel 5: out = LN(x + rw*(agg + bias)) * gamma + beta  (one wave per node)
// ---------------------------------------------------------------------
__global__ __launch_bounds__(256) void k_finalize(const float* __restrict__ x,
                                                  const float* __restrict__ agg,
                                                  const float* __restrict__ bias,
                                                  const float* __restrict__ res_w,
                                                  const float* __restrict__ gamma,
                                                  const float* __restrict__ beta,
                                                  float* __restrict__ out) {
    const int n = blockIdx.x * 8 + (threadIdx.x >> 5);
    if (n >= N_NODES) return;
    const int lane = threadIdx.x & 31;
    const float rw = res_w[0];

    const float4 xv = ((const float4*)(x   + n * DIM))[lane];
    const float4 ag = ((const float4*)(agg + n * DIM))[lane];
    const float4 bs = ((const float4*)bias)[lane];

    float4 t;
    t.x = xv.x + rw * (ag.x + bs.x);
    t.y = xv.y + rw * (ag.y + bs.y);
    t.z = xv.z + rw * (ag.z + bs.z);
    t.w = xv.w + rw * (ag.w + bs.w);

    float s  = t.x + t.y + t.z + t.w;
    float sq = t.x * t.x + t.y * t.y + t.z * t.z + t.w * t.w;
#pragma unroll
    for (int off = 16; off > 0; off >>= 1) {
        s  += __shfl_xor(s,  off, 32);
        sq += __shfl_xor(sq, off, 32);
    }
    const float mu  = s * (1.0f / DIM);
    const float var = sq * (1.0f / DIM) - mu * mu;
    const float inv = rsqrtf(var + LN_EPS);

    const float4 g  = ((const float4*)gamma)[lane];
    const float4 be = ((const float4*)beta)[lane];
    float4 o;
    o.x = (t.x - mu) * inv * g.x + be.x;
    o.y = (t.y - mu) * inv * g.y + be.y;
    o.z = (t.z - mu) * inv * g.z + be.z;
    o.w = (t.w - mu) * inv * g.w + be.w;
    ((float4*)(out + n * DIM))[lane] = o;
}

// ---------------------------------------------------------------------
extern "C" void kernel_launch(void* const* d_in, const int* in_sizes, int n_in,
                              void* d_out, int out_size, void* d_ws, size_t ws_size,
                              hipStream_t stream) {
    const float* x         = (const float*)d_in[0];
    const int*   ei        = (const int*)  d_in[1];
    const float* edge_attr = (const float*)d_in[2];
    const float* W_l       = (const float*)d_in[3];
    const float* b_l       = (const float*)d_in[4];
    const float* W_r       = (const float*)d_in[5];
    const float* b_r       = (const float*)d_in[6];
    const float* W_e       = (const float*)d_in[7];
    const float* att       = (const float*)d_in[8];
    const float* bias      = (const float*)d_in[9];
    const float* res_w     = (const float*)d_in[10];
    const float* gamma     = (const float*)d_in[11];
    const float* beta      = (const float*)d_in[12];
    float* out = (float*)d_out;

    // workspace layout (floats)
    float*    x_l     = (float*)d_ws;
    float*    x_r     = x_l + (size_t)N_NODES * DIM;
    float*    agg     = x_r + (size_t)N_NODES * DIM;
    float*    alpha   = agg + (size_t)N_NODES * DIM;        // E floats (logit -> ex, in place)
    unsigned* seg_ord = (unsigned*)(alpha + N_EDGES);
    float*    seg_sum = (float*)(seg_ord + N_NODES);

    // 0) init accumulators
    k_init<<<(N_NODES * DIM + 255) / 256, 256, 0, stream>>>(agg, seg_ord, seg_sum);

    // 1) node GEMMs: TDM-staged W + WMMA f32 16x16x4
    dim3 ggrid(N_NODES / 16, 2);
    k_gemm_xlxr<<<ggrid, 256, 0, stream>>>(x, W_l, b_l, W_r, b_r, x_l, x_r);

    // 2) edge logits + segment max
    k_edge_logits<<<N_EDGES / 8, 256, 0, stream>>>(ei, edge_attr, W_e, att,
                                                   x_l, x_r, alpha, seg_ord);

    // 3) exp + segment sum
    k_edge_exp<<<N_EDGES / 256, 256, 0, stream>>>(ei, alpha, seg_ord, seg_sum);

    // 4) normalize + scatter-add messages
    k_edge_scatter<<<N_EDGES / 8, 256, 0, stream>>>(ei, alpha, seg_sum, x_l, agg);

    // 5) residual + LayerNorm
    k_finalize<<<(N_NODES + 7) / 8, 256, 0, stream>>>(x, agg, bias, res_w,
                                                      gamma, beta, out);
}